// CustomTransformerEncoderLayer_63161789055625
// MI455X (gfx1250) — compile-verified
//
#include <hip/hip_runtime.h>

typedef float v2f __attribute__((ext_vector_type(2)));
typedef float v8f __attribute__((ext_vector_type(8)));

namespace {
constexpr int SEQ  = 2048;
constexpr int BAT  = 8;
constexpr int DIM  = 512;
constexpr int NH   = 8;
constexpr int HDIM = 64;
constexpr int N_QKV = 3 * DIM;                                  // 1536
constexpr size_t QK_STRIDE = (size_t)BAT * NH * SEQ * HDIM;     // 8,388,608 floats per q/k/v
constexpr size_t BSTRIDE   = (size_t)NH * SEQ * HDIM;           // 1,048,576 (batch stride, all layouts)
}

// D = A(16x4) * B(4x16) + C, all fp32, wave32. Numerically equivalent to fp32 FMA GEMM.
static __device__ __forceinline__ v8f wmma_f32(v2f a, v2f b, v8f c) {
  return __builtin_amdgcn_wmma_f32_16x16x4_f32(false, a, false, b, (short)0, c,
                                               false, false);
}

// ---------------------------------------------------------------------------
// Kernel 1: packed QKV projection.
//   qkv[m, n] = src[m, :] . in_proj_w[n, :] + in_proj_b[n]   (NT GEMM, K=512)
// Workspace layouts:
//   q : [B, H, S, HDIM]  (pre-scaled by hd^-0.5 = 0.125)
//   k : [B, H, S, HDIM]
//   v : [B, H, HDIM, S]  (transposed -> ctx GEMM B-fragments are b64 loads)
// Each wave: 64x64 tile = 4x4 grid of 16x16 WMMA accumulators.
// ---------------------------------------------------------------------------
__global__ __launch_bounds__(256) void qkv_proj_kernel(
    const float* __restrict__ src,   // [16384, 512] row-major
    const float* __restrict__ w,     // [1536, 512]  row-major
    const float* __restrict__ bias,  // [1536]
    float* __restrict__ qkv) {
  const int lane  = threadIdx.x & 31;
  const int wave  = threadIdx.x >> 5;
  const int mrow  = blockIdx.x * 128 + (wave >> 2) * 64;
  const int ncol  = blockIdx.y * 256 + (wave & 3) * 64;
  const int ml    = lane & 15;          // A-row / B-col / C-col lane index
  const int kk    = (lane >> 4) * 2;    // K pair base within 4-wide K step
  const int mhalf = (lane >> 4) * 8;    // C-row half select

  v8f acc[4][4];
#pragma unroll
  for (int i = 0; i < 4; ++i)
#pragma unroll
    for (int j = 0; j < 4; ++j) acc[i][j] = {};

  const float* aBase = src + (size_t)(mrow + ml) * DIM + kk;
  const float* bBase = w   + (size_t)(ncol + ml) * DIM + kk;
  for (int kb = 0; kb < DIM; kb += 4) {
    v2f af[4], bf[4];
#pragma unroll
    for (int i = 0; i < 4; ++i)
      af[i] = *(const v2f*)(aBase + (size_t)i * 16 * DIM + kb);
#pragma unroll
    for (int j = 0; j < 4; ++j)
      bf[j] = *(const v2f*)(bBase + (size_t)j * 16 * DIM + kb);
#pragma unroll
    for (int i = 0; i < 4; ++i)
#pragma unroll
      for (int j = 0; j < 4; ++j)
        acc[i][j] = wmma_f32(af[i], bf[j], acc[i][j]);
  }

  // Epilogue. A 16-wide tile (n0 = multiple of 16) never crosses a 512-
  // boundary (q/k/v) or 64-boundary (head), so `which`, `h`, `hd0` are
  // wave-uniform scalars. Along r, s is fixed and b == r, and the batch
  // stride is BSTRIDE floats in all three layouts.
#pragma unroll
  for (int j = 0; j < 4; ++j) {
    const int n0    = ncol + j * 16;       // uniform
    const int which = n0 >> 9;             // uniform: 0=q, 1=k, 2=v
    const int h     = (n0 & (DIM - 1)) >> 6;   // uniform
    const int hd0   = n0 & (HDIM - 1);         // uniform
    const float bv  = bias[n0 + ml];
    const float scl = (which == 0) ? 0.125f : 1.0f;
#pragma unroll
    for (int i = 0; i < 4; ++i) {
      const int s = (mrow + i * 16 + mhalf) >> 3;  // fixed across r (b == r)
      float* p;
      if (which == 2) {
        p = qkv + 2 * QK_STRIDE +
            ((size_t)(h * HDIM) + hd0 + ml) * SEQ + s;           // vT
      } else {
        p = qkv + (size_t)which * QK_STRIDE +
            ((size_t)h * SEQ + s) * HDIM + hd0 + ml;             // q / k
      }
#pragma unroll
      for (int r = 0; r < 8; ++r)
        p[(size_t)r * BSTRIDE] = (acc[i][j][r] + bv) * scl;
    }
  }
}

// ---------------------------------------------------------------------------
// Kernel 2: fused attention + head-mean + output projection.
// One workgroup per (16 query rows, batch). Dynamic LDS (289 KB of 320 KB WGP):
//   sc   [16][2048]  score / prob tile for current head
//   asum [16][2048]  running sum over heads (for mean attn output)
//   ctx  [16][512]   context rows (all heads), feeds out-proj without HBM trip
// ---------------------------------------------------------------------------
__global__ __launch_bounds__(256) void attn_fused_kernel(
    const float* __restrict__ qkv,    // ws from kernel 1
    const float* __restrict__ ow,     // [512, 512] row-major
    const float* __restrict__ ob,     // [512]
    float* __restrict__ out,          // [S, B, D]
    float* __restrict__ attn_out) {   // [B, S, S]
  extern __shared__ float smem[];
  float* sc   = smem;                       // 32768
  float* asum = sc + 16 * SEQ;              // 32768
  float* ctx  = asum + 16 * SEQ;            // 8192
  float* red  = ctx + 16 * DIM;             // 256
  float* rmax = red + 256;                  // 16
  float* rsum = rmax + 16;                  // 16

  const int tid   = threadIdx.x;
  const int lane  = tid & 31;
  const int wave  = tid >> 5;
  const int s0    = blockIdx.x * 16;
  const int b     = blockIdx.y;
  const int ml    = lane & 15;
  const int kk    = (lane >> 4) * 2;
  const int mhalf = (lane >> 4) * 8;

  for (int idx = tid; idx < 16 * DIM; idx += 256) ctx[idx] = 0.0f;
  for (int idx = tid; idx < 16 * SEQ; idx += 256) asum[idx] = 0.0f;
  __syncthreads();

  for (int h = 0; h < NH; ++h) {
    const float* Q  = qkv + ((size_t)(b * NH + h)) * SEQ * HDIM;
    const float* Km = Q + QK_STRIDE;
    const float* Vt = qkv + 2 * QK_STRIDE +
                      ((size_t)(b * NH + h)) * HDIM * SEQ;  // [HDIM, S]

    // ---- scores[16, 2048] = Qtile @ K^T (Q already scaled) -----------------
    const float* ap = Q + (size_t)(s0 + ml) * HDIM + kk;
    v2f aq[16];
#pragma unroll
    for (int k4 = 0; k4 < 16; ++k4) aq[k4] = *(const v2f*)(ap + k4 * 4);

    for (int t = 0; t < 16; ++t) {
      const int c0 = (wave * 16 + t) * 16;
      v8f acc = {};
      const float* bp = Km + (size_t)(c0 + ml) * HDIM + kk;
#pragma unroll
      for (int k4 = 0; k4 < 16; ++k4) {
        v2f bf = *(const v2f*)(bp + k4 * 4);
        acc = wmma_f32(aq[k4], bf, acc);
      }
#pragma unroll
      for (int r = 0; r < 8; ++r)
        sc[(mhalf + r) * SEQ + c0 + ml] = acc[r];
    }
    __syncthreads();

    // ---- softmax over 2048 cols; 16 threads cooperate per row --------------
    const int row   = tid >> 4;
    const int col16 = tid & 15;
    float m = -3.402823e38f;
    for (int c = col16; c < SEQ; c += 16) m = fmaxf(m, sc[row * SEQ + c]);
    red[row * 16 + col16] = m;
    __syncthreads();
    if (col16 == 0) {
      float mm = red[row * 16];
      for (int i = 1; i < 16; ++i) mm = fmaxf(mm, red[row * 16 + i]);
      rmax[row] = mm;
    }
    __syncthreads();
    const float rm = rmax[row];
    float ps = 0.0f;
    for (int c = col16; c < SEQ; c += 16) {
      const float e = __expf(sc[row * SEQ + c] - rm);
      sc[row * SEQ + c] = e;
      ps += e;
    }
    red[row * 16 + col16] = ps;
    __syncthreads();
    if (col16 == 0) {
      float ss = 0.0f;
      for (int i = 0; i < 16; ++i) ss += red[row * 16 + i];
      rsum[row] = ss;
    }
    __syncthreads();
    const float inv = 1.0f / rsum[row];
    for (int c = col16; c < SEQ; c += 16) {
      const float p = sc[row * SEQ + c] * inv;
      sc[row * SEQ + c] = p;
      asum[row * SEQ + c] += p;
    }
    __syncthreads();

    // ---- ctx[16, 64] = attn @ V; 4 n-tiles x 2 K-halves over 8 waves -------
    const int nb   = (wave & 3) * 16;
    const int kbeg = (wave >> 2) * (SEQ / 2);
    v8f cacc = {};
    const float* ascan = &sc[ml * SEQ + kk];
    const float* vp    = Vt + (size_t)(nb + ml) * SEQ + kk;
    for (int kb = kbeg; kb < kbeg + SEQ / 2; kb += 4) {
      v2f a  = *(const v2f*)(ascan + kb);
      v2f bf = *(const v2f*)(vp + kb);
      cacc = wmma_f32(a, bf, cacc);
    }
#pragma unroll
    for (int r = 0; r < 8; ++r)
      atomicAdd(&ctx[(mhalf + r) * DIM + h * HDIM + nb + ml], cacc[r]);
    __syncthreads();
  }

  // ---- attn_weights = mean over heads ------------------------------------
  for (int idx = tid; idx < 16 * SEQ; idx += 256) {
    const int r = idx >> 11;
    const int c = idx & (SEQ - 1);
    attn_out[(size_t)b * SEQ * SEQ + (size_t)(s0 + r) * SEQ + c] =
        asum[idx] * 0.125f;
  }

  // ---- out[16, 512] = ctx @ out_w^T + out_b (A from LDS ctx) --------------
  for (int t = 0; t < 4; ++t) {
    const int nb2 = (wave * 4 + t) * 16;
    v8f oacc = {};
    const float* wp = ow + (size_t)(nb2 + ml) * DIM + kk;
    const float* cp = &ctx[ml * DIM + kk];
    for (int kb = 0; kb < DIM; kb += 4) {
      v2f a  = *(const v2f*)(cp + kb);
      v2f bf = *(const v2f*)(wp + kb);
      oacc = wmma_f32(a, bf, oacc);
    }
    const float bv = ob[nb2 + ml];
#pragma unroll
    for (int r = 0; r < 8; ++r) {
      const int srow = s0 + mhalf + r;
      out[((size_t)srow * BAT + b) * DIM + nb2 + ml] = oacc[r] + bv;
    }
  }
}

extern "C" void kernel_launch(void* const* d_in, const int* in_sizes, int n_in,
                              void* d_out, int out_size, void* d_ws,
                              size_t ws_size, hipStream_t stream) {
  (void)in_sizes; (void)n_in; (void)out_size; (void)ws_size;
  const float* src  = (const float*)d_in[0];
  const float* in_w = (const float*)d_in[1];
  const float* in_b = (const float*)d_in[2];
  const float* ow   = (const float*)d_in[3];
  const float* ob   = (const float*)d_in[4];
  float* out  = (float*)d_out;
  float* attn = (float*)d_out + (size_t)SEQ * BAT * DIM;
  float* ws   = (float*)d_ws;   // q|k|vT, 3 * 8,388,608 floats = ~100 MB

  dim3 gA((SEQ * BAT) / 128, N_QKV / 256);   // 128 x 6
  qkv_proj_kernel<<<gA, dim3(256), 0, stream>>>(src, in_w, in_b, ws);

  dim3 gB(SEQ / 16, BAT);                    // 128 x 8
  const size_t shmem =
      (size_t)(2 * 16 * SEQ + 16 * DIM + 256 + 32) * sizeof(float);  // 296,064 B
  attn_fused_kernel<<<gB, dim3(256), shmem, stream>>>(ws, ow, ob, out, attn);
}